// RKDLoss_22041772163495
// MI455X (gfx1250) — compile-verified
//
#include <hip/hip_runtime.h>
#include <hip/hip_bf16.h>

// ---------------------------------------------------------------------------
// RKD loss (dist + angle + hard CE) for B=384, C=512 f32 inputs on gfx1250.
//
// Key optimization: angle gram (f_j-f_i)·(f_k-f_i)/(r_ij r_ik) is computed
// from the Gram matrix G=F F^T via
//     num = G[j,k] - G[i,j] - G[i,k] + sq[i],   r_ij^2 = sq[i]+sq[j]-2G[i,j]
// so the only GEMMs are two 384x384x512 f32 GEMMs done with
// V_WMMA_F32_16X16X4_F32 (full f32 precision; needed because sq_i+sq_j-2g
// is a catastrophic-cancellation expression).
// ---------------------------------------------------------------------------

#define B_DIM 384
#define C_DIM 512
#define GSZ   (B_DIM * B_DIM)   // 147456
#define EPSF  1e-12f

typedef __attribute__((ext_vector_type(2))) float v2f;
typedef __attribute__((ext_vector_type(8))) float v8f;

// ---------------- workspace layout (floats) ----------------
// gram_s  : [0,       GSZ)
// gram_t  : [GSZ,    2GSZ)
// d_s     : [2GSZ,   3GSZ)
// d_t     : [3GSZ,   4GSZ)
// inv_s   : [4GSZ,   5GSZ)
// inv_t   : [5GSZ,   6GSZ)
// partA   : [6GSZ,   6GSZ+2304)        576 blocks * 4 floats
// partAng : [.. +2304, +2304+9216)     384*24 blocks
// partCE  : [.. +9216, +9216+384)
#define OFF_GS   0
#define OFF_GT   (GSZ)
#define OFF_DS   (2 * GSZ)
#define OFF_DT   (3 * GSZ)
#define OFF_IVS  (4 * GSZ)
#define OFF_IVT  (5 * GSZ)
#define OFF_PA   (6 * GSZ)
#define N_PA_BLK 576
#define OFF_PANG (OFF_PA + 4 * N_PA_BLK)
#define N_ANG_BLK (B_DIM * 24)
#define OFF_PCE  (OFF_PANG + N_ANG_BLK)

__device__ __forceinline__ float block_reduce_256(float v, float* sbuf) {
    const int t = threadIdx.x;
    sbuf[t] = v;
    __syncthreads();
#pragma unroll
    for (int s = 128; s > 0; s >>= 1) {
        if (t < s) sbuf[t] += sbuf[t + s];
        __syncthreads();
    }
    float r = sbuf[0];
    __syncthreads();
    return r;
}

__device__ __forceinline__ float smooth_l1(float x, float y) {
    float a = fabsf(x - y);
    return (a < 1.0f) ? 0.5f * a * a : a - 0.5f;
}

// --------------------------------------------------------------------------
// Kernel 1: Gram matrices via V_WMMA_F32_16X16X4_F32.
// One wave per 16x16 output tile; grid = (24*24 tiles, 2 matrices).
// f32 WMMA operand layout (ISA 7.12.2):
//   A 16x4 : lanes 0-15 row M=lane, VGPR c -> K=c;     lanes 16-31 -> K=c+2
//   B 4x16 : lanes 0-15 col N=lane, VGPR c -> K=c;     lanes 16-31 -> K=c+2
//   C 16x16: VGPR v, lanes 0-15 -> M=v, lanes 16-31 -> M=v+8, N=lane&15
// --------------------------------------------------------------------------
__global__ __launch_bounds__(32) void rkd_gram_wmma(const float* __restrict__ fs,
                                                    const float* __restrict__ ft,
                                                    float* __restrict__ ws) {
    const int tile  = blockIdx.x;           // 0..575
    const int which = blockIdx.y;           // 0 = student, 1 = teacher
    const float* f = which ? ft : fs;
    float* g       = ws + (which ? OFF_GT : OFF_GS);

    const int tr = (tile / 24) * 16;        // row base
    const int tc = (tile % 24) * 16;        // col base
    const int lane = threadIdx.x;           // 0..31
    const int half = lane >> 4;             // 0 or 1
    const int m    = lane & 15;

    const float* arow = f + (tr + m) * C_DIM + 2 * half;  // A: rows of tile-row
    const float* brow = f + (tc + m) * C_DIM + 2 * half;  // B: rows of tile-col (=F^T cols)

    v8f acc = {};
#pragma unroll 4
    for (int k = 0; k < C_DIM; k += 4) {
        v2f a = *(const v2f*)(arow + k);
        v2f b = *(const v2f*)(brow + k);
        acc = __builtin_amdgcn_wmma_f32_16x16x4_f32(
            /*neg_a=*/false, a, /*neg_b=*/false, b,
            /*c_mod=*/(short)0, acc, /*reuse_a=*/false, /*reuse_b=*/false);
    }

#pragma unroll
    for (int v = 0; v < 8; ++v) {
        g[(tr + v + 8 * half) * B_DIM + (tc + m)] = acc[v];
    }
}

// --------------------------------------------------------------------------
// Kernel 2: per-pair distances d (pdist, diag zeroed) and inverse norms
// 1/max(r,eps) (with inv_ii forced to 0 so e_ii = 0 exactly), plus
// per-block partial sums of d and positive counts for the mean.
// Grid: 576 blocks x 256 threads over 147456 pairs.
// --------------------------------------------------------------------------
__global__ __launch_bounds__(256) void rkd_dist_inv(float* __restrict__ ws) {
    __shared__ float sbuf[256];
    const int idx = blockIdx.x * 256 + threadIdx.x;
    const int i = idx / B_DIM;
    const int j = idx % B_DIM;

    const float* gram_s = ws + OFF_GS;
    const float* gram_t = ws + OFF_GT;

    const float gs = gram_s[idx];
    const float gt = gram_t[idx];
    const float r2s = gram_s[i * (B_DIM + 1)] + gram_s[j * (B_DIM + 1)] - 2.0f * gs;
    const float r2t = gram_t[i * (B_DIM + 1)] + gram_t[j * (B_DIM + 1)] - 2.0f * gt;

    float ds, dt, ivs, ivt;
    if (i == j) {
        ds = dt = 0.0f;
        ivs = ivt = 0.0f;     // forces angle terms with j==i or k==i to zero
    } else {
        ds = sqrtf(fmaxf(r2s, EPSF));
        dt = sqrtf(fmaxf(r2t, EPSF));
        ivs = 1.0f / fmaxf(sqrtf(fmaxf(r2s, 0.0f)), EPSF);
        ivt = 1.0f / fmaxf(sqrtf(fmaxf(r2t, 0.0f)), EPSF);
    }
    ws[OFF_DS + idx]  = ds;
    ws[OFF_DT + idx]  = dt;
    ws[OFF_IVS + idx] = ivs;
    ws[OFF_IVT + idx] = ivt;

    const float sum_ds = block_reduce_256(ds, sbuf);
    const float sum_dt = block_reduce_256(dt, sbuf);
    const float cnt_s  = block_reduce_256(ds > 0.0f ? 1.0f : 0.0f, sbuf);
    const float cnt_t  = block_reduce_256(dt > 0.0f ? 1.0f : 0.0f, sbuf);
    if (threadIdx.x == 0) {
        float* pa = ws + OFF_PA + blockIdx.x * 4;
        pa[0] = sum_ds; pa[1] = sum_dt; pa[2] = cnt_s; pa[3] = cnt_t;
    }
}

// --------------------------------------------------------------------------
// Kernel 3: angle-gram smooth-L1 partial sums over the [B,B,B] tensor,
// using the Gram-expansion identity. Block = (anchor i, tile of 16 j's),
// 256 threads sweep 16*384 (j,k) elements. Anchor row of G and inv cached
// in LDS; G[j,k] streamed from L2.
// --------------------------------------------------------------------------
__global__ __launch_bounds__(256) void rkd_angle(const float* __restrict__ ws_ro,
                                                 float* __restrict__ ws) {
    __shared__ float row_gs[B_DIM];
    __shared__ float row_gt[B_DIM];
    __shared__ float row_ivs[B_DIM];
    __shared__ float row_ivt[B_DIM];
    __shared__ float sbuf[256];

    const int i  = blockIdx.x;      // anchor
    const int jt = blockIdx.y;      // j tile (16 wide)
    const int t  = threadIdx.x;

    const float* gram_s = ws_ro + OFF_GS;
    const float* gram_t = ws_ro + OFF_GT;
    const float* inv_s  = ws_ro + OFF_IVS;
    const float* inv_t  = ws_ro + OFF_IVT;

    for (int c = t; c < B_DIM; c += 256) {
        row_gs[c]  = gram_s[i * B_DIM + c];
        row_gt[c]  = gram_t[i * B_DIM + c];
        row_ivs[c] = inv_s[i * B_DIM + c];
        row_ivt[c] = inv_t[i * B_DIM + c];
    }
    __syncthreads();

    const float sqs_i = row_gs[i];
    const float sqt_i = row_gt[i];

    float acc = 0.0f;
    // 16 j's x 384 k's = 6144 elements, 24 per thread
    for (int e = t; e < 16 * B_DIM; e += 256) {
        const int jj = e / B_DIM;
        const int k  = e % B_DIM;
        const int j  = jt * 16 + jj;

        const float gs_jk = gram_s[j * B_DIM + k];
        const float gt_jk = gram_t[j * B_DIM + k];

        const float num_s = gs_jk - row_gs[j] - row_gs[k] + sqs_i;
        const float num_t = gt_jk - row_gt[j] - row_gt[k] + sqt_i;
        const float gs_v  = num_s * row_ivs[j] * row_ivs[k];
        const float gt_v  = num_t * row_ivt[j] * row_ivt[k];
        acc += smooth_l1(gs_v, gt_v);
    }

    const float total = block_reduce_256(acc, sbuf);
    if (t == 0) ws[OFF_PANG + i * 24 + jt] = total;
}

// --------------------------------------------------------------------------
// Kernel 4: hard CE partials: partCE[row] = logsumexp(feat_s[row]) - x[label].
// One block of 256 threads per row (512 cols -> 2 per thread).
// --------------------------------------------------------------------------
__global__ __launch_bounds__(256) void rkd_ce(const float* __restrict__ fs,
                                              const int* __restrict__ labels,
                                              float* __restrict__ ws) {
    __shared__ float sbuf[256];
    const int row = blockIdx.x;
    const int t   = threadIdx.x;
    const float* x = fs + row * C_DIM;

    float x0 = x[t];
    float x1 = x[t + 256];
    float m = fmaxf(x0, x1);
    // max reduce
    sbuf[t] = m;
    __syncthreads();
#pragma unroll
    for (int s = 128; s > 0; s >>= 1) {
        if (t < s) sbuf[t] = fmaxf(sbuf[t], sbuf[t + s]);
        __syncthreads();
    }
    const float mx = sbuf[0];
    __syncthreads();

    const float se = block_reduce_256(__expf(x0 - mx) + __expf(x1 - mx), sbuf);
    if (t == 0) {
        const int lbl = labels[row];
        ws[OFF_PCE + row] = (mx + logf(se)) - x[lbl];
    }
}

// --------------------------------------------------------------------------
// Kernel 5: finalize. Single block:
//   (a) reduce partA -> mean_pos for s and t
//   (b) sweep 147456 pairs for dist smooth-L1 (fixed order -> deterministic)
//   (c) reduce angle partials, (d) reduce CE partials, write scalar out.
// --------------------------------------------------------------------------
__global__ __launch_bounds__(256) void rkd_finalize(const float* __restrict__ ws,
                                                    float* __restrict__ out) {
    __shared__ float sbuf[256];
    const int t = threadIdx.x;

    // (a) means
    float a0 = 0.f, a1 = 0.f, a2 = 0.f, a3 = 0.f;
    for (int b = t; b < N_PA_BLK; b += 256) {
        const float* pa = ws + OFF_PA + b * 4;
        a0 += pa[0]; a1 += pa[1]; a2 += pa[2]; a3 += pa[3];
    }
    const float sum_ds = block_reduce_256(a0, sbuf);
    const float sum_dt = block_reduce_256(a1, sbuf);
    const float cnt_s  = block_reduce_256(a2, sbuf);
    const float cnt_t  = block_reduce_256(a3, sbuf);
    const float inv_mean_s = fmaxf(cnt_s, 1.0f) / sum_ds;
    const float inv_mean_t = fmaxf(cnt_t, 1.0f) / sum_dt;

    // (b) dist smooth-L1
    const float* d_s = ws + OFF_DS;
    const float* d_t = ws + OFF_DT;
    float dacc = 0.0f;
    for (int idx = t; idx < GSZ; idx += 256) {
        dacc += smooth_l1(d_s[idx] * inv_mean_s, d_t[idx] * inv_mean_t);
    }
    const float dist_loss = block_reduce_256(dacc, sbuf) / (float)GSZ;

    // (c) angle
    float aacc = 0.0f;
    for (int b = t; b < N_ANG_BLK; b += 256) aacc += ws[OFF_PANG + b];
    const float angle_loss =
        block_reduce_256(aacc, sbuf) / ((float)B_DIM * (float)B_DIM * (float)B_DIM);

    // (d) CE
    float cacc = 0.0f;
    for (int b = t; b < B_DIM; b += 256) cacc += ws[OFF_PCE + b];
    const float ce = block_reduce_256(cacc, sbuf) / (float)B_DIM;

    if (t == 0) out[0] = ce + dist_loss + angle_loss;
}

// --------------------------------------------------------------------------
extern "C" void kernel_launch(void* const* d_in, const int* in_sizes, int n_in,
                              void* d_out, int out_size, void* d_ws, size_t ws_size,
                              hipStream_t stream) {
    (void)in_sizes; (void)n_in; (void)out_size; (void)ws_size;
    const float* feat_s = (const float*)d_in[0];
    const float* feat_t = (const float*)d_in[1];
    const int*   labels = (const int*)d_in[2];
    float* out = (float*)d_out;
    float* ws  = (float*)d_ws;

    // 1) Gram matrices via WMMA f32
    rkd_gram_wmma<<<dim3(576, 2), dim3(32), 0, stream>>>(feat_s, feat_t, ws);
    // 2) distances / inverse norms + partial sums
    rkd_dist_inv<<<dim3(N_PA_BLK), dim3(256), 0, stream>>>(ws);
    // 3) angle-gram smooth-L1 partials over [B,B,B]
    rkd_angle<<<dim3(B_DIM, 24), dim3(256), 0, stream>>>(ws, ws);
    // 4) hard CE partials
    rkd_ce<<<dim3(B_DIM), dim3(256), 0, stream>>>(feat_s, labels, ws);
    // 5) finalize scalar
    rkd_finalize<<<dim3(1), dim3(256), 0, stream>>>(ws, out);
}